// CrossDomainAttention_2370821948241
// MI455X (gfx1250) — compile-verified
//
#include <hip/hip_runtime.h>
#include <hip/hip_bf16.h>

typedef __attribute__((ext_vector_type(2))) float v2f;
typedef __attribute__((ext_vector_type(8))) float v8f;

#define NTOK 131072     // D*H*W = 32*64*64
#define NCH  128        // C = E
#define NB   4          // batch
#define NHD  4          // heads
#define HDM  32         // head dim
#define SCALE 0.17677669529663687f  // 1/sqrt(32)

// -------- workspace layout (bytes) --------
#define SCORES_OFF   0u            // B*4*N  f32  = 8388608 B
#define WEXP_OFF     8388608u      // B*N float4  = 8388608 B
#define U_OFF        16777216u     // B*16*128 f32 = 32768 B
#define QB_OFF       16809984u     // 16 f32
#define MAX_OFF      16810048u     // 16 f32
#define ZPART_OFF    16810112u     // B*64*4 f32 = 4096 B
#define TPART_OFF    16814208u     // B*64*4*128 f32 = 524288 B
#define LN_OFF       17338496u     // B*128 f32

// ---------------------------------------------------------------------------
// Kernel 0: tiny setup. q = emb[dom]@Wq_proj.T+b ; qh = q@Wq.T+bq ;
// U[b,h,:] = scale * qh_h @ wk_h ; qk_bias = scale * qh_h . bk_h.
// Rows 4..15 of U zero-padded (WMMA A-tile is 16 rows).
// ---------------------------------------------------------------------------
__global__ __launch_bounds__(256) void k_setup(
    const int* __restrict__ dom, const float* __restrict__ emb,
    const float* __restrict__ qpw, const float* __restrict__ qpb,
    const float* __restrict__ wq, const float* __restrict__ bq,
    const float* __restrict__ wk, const float* __restrict__ bk,
    float* __restrict__ U, float* __restrict__ qb) {
  __shared__ float sQ[128], sQh[128];
  const int tid = threadIdx.x;
  for (int b = 0; b < NB; ++b) {
    const int d = dom[b];
    if (tid < 128) {
      float s = qpb[tid];
      for (int c = 0; c < 128; ++c) s += emb[d * 128 + c] * qpw[tid * 128 + c];
      sQ[tid] = s;
    }
    __syncthreads();
    if (tid < 128) {
      float s = bq[tid];
      for (int c = 0; c < 128; ++c) s += sQ[c] * wq[tid * 128 + c];
      sQh[tid] = s;
    }
    __syncthreads();
    if (tid < 128) {
      const int c = tid;
      for (int h = 0; h < NHD; ++h) {
        float u = 0.f;
        for (int dd = 0; dd < HDM; ++dd)
          u += sQh[h * HDM + dd] * wk[(h * HDM + dd) * 128 + c];
        U[b * 2048 + h * 128 + c] = u * SCALE;
      }
      for (int r = NHD; r < 16; ++r) U[b * 2048 + r * 128 + c] = 0.f;
    }
    if (tid < NHD) {
      float s = 0.f;
      for (int dd = 0; dd < HDM; ++dd) s += sQh[tid * HDM + dd] * bk[tid * HDM + dd];
      qb[b * 4 + tid] = s * SCALE;
    }
    __syncthreads();
  }
}

// ---------------------------------------------------------------------------
// Kernel 1: scores = U(16x128) x X(128xN) via V_WMMA_F32_16X16X4_F32.
// Each wave owns 16-token tiles; K swept 128 in chunks of 4 (32 WMMAs/tile).
// A-tile (U) from LDS; B-tile (x) loaded per-lane, coalesced along tokens.
// ---------------------------------------------------------------------------
__global__ __launch_bounds__(256) void k_scores(
    const float* __restrict__ x, const float* __restrict__ U,
    const float* __restrict__ qb, float* __restrict__ scores) {
  __shared__ float sU[16 * 128];
  const int b = blockIdx.y;
  const int tid = threadIdx.x;
  for (int i = tid; i < 2048; i += 256) sU[i] = U[b * 2048 + i];
  __syncthreads();

  const float* xb = x + (size_t)b * NCH * NTOK;
  float* sb = scores + (size_t)b * 4 * NTOK;
  const int lane = tid & 31;
  const int wave = tid >> 5;
  const int m = lane & 15;            // row (A) / token-within-tile (B)
  const int koff = (lane >> 4) << 1;  // half-wave K split: 0 or 2
  const float qb0 = qb[b * 4 + 0], qb1 = qb[b * 4 + 1];
  const float qb2 = qb[b * 4 + 2], qb3 = qb[b * 4 + 3];

  const int wid = blockIdx.x * 8 + wave;          // 2048 waves per batch
  for (int tile = wid; tile < NTOK / 16; tile += 2048) {
    const int n = tile * 16 + m;
    v8f acc = {};
#pragma unroll 4
    for (int kc = 0; kc < 128; kc += 4) {
      v2f a, bm;
      a.x = sU[m * 128 + kc + koff];
      a.y = sU[m * 128 + kc + koff + 1];
      bm.x = xb[(size_t)(kc + koff) * NTOK + n];
      bm.y = xb[(size_t)(kc + koff + 1) * NTOK + n];
      acc = __builtin_amdgcn_wmma_f32_16x16x4_f32(false, a, false, bm,
                                                  (short)0, acc, false, false);
    }
    if (lane < 16) {  // rows 0..3 = heads, live in acc[0..3] lanes 0..15
      sb[0 * NTOK + n] = acc[0] + qb0;
      sb[1 * NTOK + n] = acc[1] + qb1;
      sb[2 * NTOK + n] = acc[2] + qb2;
      sb[3 * NTOK + n] = acc[3] + qb3;
    }
  }
}

// ---------------------------------------------------------------------------
// Kernel 2: per-(b,h) max over N (scores are L2-resident, 8 MB total).
// ---------------------------------------------------------------------------
__global__ __launch_bounds__(256) void k_max(const float* __restrict__ scores,
                                             float* __restrict__ maxbuf) {
  __shared__ float red[256];
  const int bh = blockIdx.x;
  const float4* s4 = (const float4*)(scores + (size_t)bh * NTOK);
  float mm = -3.4e38f;
  for (int i = threadIdx.x; i < NTOK / 4; i += 256) {
    float4 v = s4[i];
    mm = fmaxf(mm, fmaxf(fmaxf(v.x, v.y), fmaxf(v.z, v.w)));
  }
  red[threadIdx.x] = mm;
  __syncthreads();
  for (int s = 128; s > 0; s >>= 1) {
    if (threadIdx.x < s) red[threadIdx.x] = fmaxf(red[threadIdx.x], red[threadIdx.x + s]);
    __syncthreads();
  }
  if (threadIdx.x == 0) maxbuf[bh] = red[0];
}

// ---------------------------------------------------------------------------
// Kernel 3a: wexp[b,n,0..3] = exp(score - max); deterministic Z partials
// per block (no float atomics).
// ---------------------------------------------------------------------------
__global__ __launch_bounds__(256) void k_exp(
    const float* __restrict__ scores, const float* __restrict__ maxbuf,
    float4* __restrict__ wexp4, float* __restrict__ zpart) {
  const int b = blockIdx.y;
  const int tid = threadIdx.x;
  const float* sb = scores + (size_t)b * 4 * NTOK;
  const float m0 = maxbuf[b * 4 + 0], m1 = maxbuf[b * 4 + 1];
  const float m2 = maxbuf[b * 4 + 2], m3 = maxbuf[b * 4 + 3];
  float z[4] = {0.f, 0.f, 0.f, 0.f};
  const int base = blockIdx.x * 2048;
  for (int k = 0; k < 8; ++k) {
    const int n = base + k * 256 + tid;
    float e0 = __expf(sb[0 * NTOK + n] - m0);
    float e1 = __expf(sb[1 * NTOK + n] - m1);
    float e2 = __expf(sb[2 * NTOK + n] - m2);
    float e3 = __expf(sb[3 * NTOK + n] - m3);
    wexp4[(size_t)b * NTOK + n] = make_float4(e0, e1, e2, e3);
    z[0] += e0; z[1] += e1; z[2] += e2; z[3] += e3;
  }
  __shared__ float red[256];
  for (int h = 0; h < 4; ++h) {
    red[tid] = z[h];
    __syncthreads();
    for (int s = 128; s > 0; s >>= 1) {
      if (tid < s) red[tid] += red[tid + s];
      __syncthreads();
    }
    if (tid == 0) zpart[(b * 64 + blockIdx.x) * 4 + h] = red[0];
    __syncthreads();
  }
}

// ---------------------------------------------------------------------------
// Kernel 3b: t_partial[b,blk,h,c] = sum over 2048-token tile of w[h,n]*x[c,n].
// Weights staged in LDS (float4 -> ds_load_b128); x streamed coalesced.
// ---------------------------------------------------------------------------
__global__ __launch_bounds__(256) void k_wsum(
    const float* __restrict__ x, const float4* __restrict__ wexp4,
    float* __restrict__ tpart) {
  __shared__ float4 sW[2048];  // 32 KB
  const int b = blockIdx.y;
  const int tid = threadIdx.x;
  const int n0 = blockIdx.x * 2048;
  for (int i = tid; i < 2048; i += 256) sW[i] = wexp4[(size_t)b * NTOK + n0 + i];
  __syncthreads();

  const float* xb = x + (size_t)b * NCH * NTOK;
  const int lane = tid & 31, wave = tid >> 5;
  for (int g = 0; g < 16; ++g) {
    const int c = wave * 16 + g;
    const float* xc = xb + (size_t)c * NTOK + n0;
    float a0 = 0.f, a1 = 0.f, a2 = 0.f, a3 = 0.f;
    for (int i = lane; i < 2048; i += 32) {
      const float xv = xc[i];
      const float4 w = sW[i];
      a0 += w.x * xv; a1 += w.y * xv; a2 += w.z * xv; a3 += w.w * xv;
    }
    for (int off = 16; off >= 1; off >>= 1) {
      a0 += __shfl_xor(a0, off);
      a1 += __shfl_xor(a1, off);
      a2 += __shfl_xor(a2, off);
      a3 += __shfl_xor(a3, off);
    }
    if (lane == 0) {
      const int pb = (b * 64 + blockIdx.x) * 4;
      tpart[(pb + 0) * 128 + c] = a0;
      tpart[(pb + 1) * 128 + c] = a1;
      tpart[(pb + 2) * 128 + c] = a2;
      tpart[(pb + 3) * 128 + c] = a3;
    }
  }
}

// ---------------------------------------------------------------------------
// Kernel 4: deterministic fuse-up. Z & t reduction, ctx = wv_h @ t + bv,
// out = wo @ ctx + bo, LayerNorm -> ln[b,128].
// ---------------------------------------------------------------------------
__global__ __launch_bounds__(256) void k_head(
    const float* __restrict__ zpart, const float* __restrict__ tpart,
    const float* __restrict__ wv_, const float* __restrict__ bv,
    const float* __restrict__ wo, const float* __restrict__ bo,
    const float* __restrict__ lng, const float* __restrict__ lnb,
    float* __restrict__ lnout) {
  __shared__ float sT[512], sCtx[128], sOut[128], red[128], sZ[16];
  const int tid = threadIdx.x;
  if (tid < 16) {
    float s = 0.f;
    const int b = tid >> 2, h = tid & 3;
    for (int k = 0; k < 64; ++k) s += zpart[(b * 64 + k) * 4 + h];
    sZ[tid] = s;
  }
  __syncthreads();
  for (int b = 0; b < NB; ++b) {
    for (int i = tid; i < 512; i += 256) {
      const int h = i >> 7, c = i & 127;
      float s = 0.f;
      for (int k = 0; k < 64; ++k) s += tpart[((b * 64 + k) * 4 + h) * 128 + c];
      sT[i] = s / sZ[b * 4 + h];
    }
    __syncthreads();
    if (tid < 128) {
      const int h = tid >> 5;
      float s = bv[tid];
      for (int c = 0; c < 128; ++c) s += wv_[tid * 128 + c] * sT[h * 128 + c];
      sCtx[tid] = s;
    }
    __syncthreads();
    if (tid < 128) {
      float o = bo[tid];
      for (int j = 0; j < 128; ++j) o += wo[tid * 128 + j] * sCtx[j];
      sOut[tid] = o;
    }
    __syncthreads();
    if (tid < 128) red[tid] = sOut[tid];
    __syncthreads();
    for (int s = 64; s > 0; s >>= 1) {
      if (tid < s) red[tid] += red[tid + s];
      __syncthreads();
    }
    const float mu = red[0] * (1.0f / 128.0f);
    __syncthreads();
    if (tid < 128) { const float d = sOut[tid] - mu; red[tid] = d * d; }
    __syncthreads();
    for (int s = 64; s > 0; s >>= 1) {
      if (tid < s) red[tid] += red[tid + s];
      __syncthreads();
    }
    const float var = red[0] * (1.0f / 128.0f);
    __syncthreads();
    if (tid < 128)
      lnout[b * 128 + tid] =
          (sOut[tid] - mu) * rsqrtf(var + 1e-5f) * lng[tid] + lnb[tid];
    __syncthreads();
  }
}

// ---------------------------------------------------------------------------
// Kernel 5: out[b,c,n] = x[b,c,n] + ln[b,c]  (float4, 537 MB traffic).
// ---------------------------------------------------------------------------
__global__ __launch_bounds__(256) void k_resid(
    const float4* __restrict__ x4, const float* __restrict__ lnbuf,
    float4* __restrict__ o4) {
  const int total4 = NB * NCH * (NTOK / 4);  // 16,777,216
  const int stride = gridDim.x * 256;
  for (int i = blockIdx.x * 256 + threadIdx.x; i < total4; i += stride) {
    const int bc = i >> 15;         // / (NTOK/4)
    const float lv = lnbuf[((bc >> 7) * 128) + (bc & 127)];
    float4 v = x4[i];
    v.x += lv; v.y += lv; v.z += lv; v.w += lv;
    o4[i] = v;
  }
}

extern "C" void kernel_launch(void* const* d_in, const int* in_sizes, int n_in,
                              void* d_out, int out_size, void* d_ws, size_t ws_size,
                              hipStream_t stream) {
  const float* x   = (const float*)d_in[0];
  const int*   dom = (const int*)d_in[1];
  const float* emb = (const float*)d_in[2];
  const float* qpw = (const float*)d_in[3];
  const float* qpb = (const float*)d_in[4];
  const float* wq  = (const float*)d_in[5];
  const float* bq  = (const float*)d_in[6];
  const float* wk  = (const float*)d_in[7];
  const float* bk  = (const float*)d_in[8];
  const float* wv_ = (const float*)d_in[9];
  const float* bv  = (const float*)d_in[10];
  const float* wo  = (const float*)d_in[11];
  const float* bo  = (const float*)d_in[12];
  const float* lng = (const float*)d_in[13];
  const float* lnb = (const float*)d_in[14];
  float* out = (float*)d_out;

  char* w = (char*)d_ws;
  float*  scores = (float*)(w + SCORES_OFF);
  float4* wexp4  = (float4*)(w + WEXP_OFF);
  float*  U      = (float*)(w + U_OFF);
  float*  qb     = (float*)(w + QB_OFF);
  float*  maxbuf = (float*)(w + MAX_OFF);
  float*  zpart  = (float*)(w + ZPART_OFF);
  float*  tpart  = (float*)(w + TPART_OFF);
  float*  lnout  = (float*)(w + LN_OFF);

  k_setup<<<1, 256, 0, stream>>>(dom, emb, qpw, qpb, wq, bq, wk, bk, U, qb);
  k_scores<<<dim3(256, NB), 256, 0, stream>>>(x, U, qb, scores);
  k_max<<<NB * NHD, 256, 0, stream>>>(scores, maxbuf);
  k_exp<<<dim3(64, NB), 256, 0, stream>>>(scores, maxbuf, wexp4, zpart);
  k_wsum<<<dim3(64, NB), 256, 0, stream>>>(x, wexp4, tpart);
  k_head<<<1, 256, 0, stream>>>(zpart, tpart, wv_, bv, wo, bo, lng, lnb, lnout);
  k_resid<<<32768, 256, 0, stream>>>((const float4*)x, lnout, (float4*)out);
}